// TopKActivation_7267084665347
// MI455X (gfx1250) — compile-verified
//
#include <hip/hip_runtime.h>
#include <stdint.h>

#define ROW_LEN 8192
#define KSEL    128
#define BLOCK   256
#define CHUNKS_PER_THREAD (ROW_LEN / (BLOCK * 4))   // 8 x 16B chunks per thread

typedef float v4f __attribute__((ext_vector_type(4)));

__global__ __launch_bounds__(BLOCK)
void topk_relu_rows(const float* __restrict__ x, float* __restrict__ out) {
    __shared__ __align__(16) float row[ROW_LEN];    // 32 KB staged row
    __shared__ unsigned int hist[256];              // radix histogram
    __shared__ unsigned int scanA[256];             // suffix-scan ping
    __shared__ unsigned int scanB[256];             // suffix-scan pong
    __shared__ unsigned int s_krem;
    __shared__ unsigned int s_prefix;
    __shared__ unsigned int s_done;
    __shared__ int          s_sel;

    const int tid = threadIdx.x;
    const long long rbase = (long long)blockIdx.x * ROW_LEN;
    const float* grow = x + rbase;
    float*       gout = out + rbase;

    // ---- Stage row into LDS via CDNA5 async global->LDS DMA (ASYNCcnt path) ----
    // Input is streamed once: mark non-temporal so it doesn't churn L2.
    {
        unsigned int lds_base = (unsigned int)(size_t)(&row[0]);
#pragma unroll
        for (int j = 0; j < CHUNKS_PER_THREAD; ++j) {
            const int chunk = tid + j * BLOCK;                 // 16-byte chunk id
            unsigned int loff = lds_base + (unsigned int)(chunk * 16);
            const float* gaddr = grow + chunk * 4;
            asm volatile("global_load_async_to_lds_b128 %0, %1, off th:TH_LOAD_NT"
                         :: "v"(loff), "v"(gaddr)
                         : "memory");
        }
        asm volatile("s_wait_asynccnt 0" ::: "memory");
    }
    __syncthreads();

    if (tid == 0) { s_krem = KSEL; s_prefix = 0u; s_done = 0u; }
    __syncthreads();

    // ---- 4-round radix select (8-bit digits, MSB first) on positive fp32 bits ----
    unsigned int prefix = 0u;
    for (int d = 3; d >= 0; --d) {
        hist[tid] = 0u;              // BLOCK == 256 bins
        if (tid == 0) s_sel = -1;
        __syncthreads();

        const int shift = d * 8;
#pragma unroll
        for (int j = 0; j < CHUNKS_PER_THREAD; ++j) {
            const int i4 = tid + j * BLOCK;
            v4f v = ((const v4f*)row)[i4];
#pragma unroll
            for (int q = 0; q < 4; ++q) {
                const float f = v[q];
                if (f > 0.0f) {
                    const unsigned int b = __float_as_uint(f);
                    if (d == 3 || (b >> (shift + 8)) == prefix) {
                        atomicAdd(&hist[(b >> shift) & 0xFFu], 1u);
                    }
                }
            }
        }
        __syncthreads();

        // Parallel suffix sum over the 256 bins: src[t] = sum_{b>=t} hist[b]
        scanA[tid] = hist[tid];
        __syncthreads();
        unsigned int* src = scanA;
        unsigned int* dst = scanB;
#pragma unroll
        for (int off = 1; off < 256; off <<= 1) {
            unsigned int v = src[tid];
            if (tid + off < 256) v += src[tid + off];
            dst[tid] = v;
            __syncthreads();
            unsigned int* t = src; src = dst; dst = t;
        }

        // Selected digit = max bin with suffix[bin] >= k_rem
        const unsigned int kr = s_krem;
        if (src[tid] >= kr) atomicMax(&s_sel, tid);
        __syncthreads();

        if (tid == 0) {
            const int sel = s_sel;
            if (sel < 0) {
                // fewer than K positive entries in this row: keep all positives
                s_done = 1u;
                s_prefix = 0u;
            } else {
                const unsigned int above = (sel < 255) ? src[sel + 1] : 0u;
                s_krem   = kr - above;
                s_prefix = (s_prefix << 8) | (unsigned int)sel;
            }
        }
        __syncthreads();
        if (s_done) break;           // uniform across block
        prefix = s_prefix;
    }

    const unsigned int T = s_prefix; // 128th-largest positive's bit pattern (or 0)

    // ---- Masked write-back: streamed once -> non-temporal stores ----
#pragma unroll
    for (int j = 0; j < CHUNKS_PER_THREAD; ++j) {
        const int i4 = tid + j * BLOCK;
        v4f v = ((const v4f*)row)[i4];
#pragma unroll
        for (int q = 0; q < 4; ++q) {
            const float f = v[q];
            v[q] = (f > 0.0f && __float_as_uint(f) >= T) ? f : 0.0f;
        }
        __builtin_nontemporal_store(v, (v4f*)gout + i4);
    }
}

extern "C" void kernel_launch(void* const* d_in, const int* in_sizes, int n_in,
                              void* d_out, int out_size, void* d_ws, size_t ws_size,
                              hipStream_t stream) {
    (void)n_in; (void)d_ws; (void)ws_size; (void)out_size;
    const float* x = (const float*)d_in[0];
    float* out = (float*)d_out;
    const int rows = in_sizes[0] / ROW_LEN;   // 8192
    topk_relu_rows<<<dim3(rows), dim3(BLOCK), 0, stream>>>(x, out);
}